// SM_GCN_17136919511825
// MI455X (gfx1250) — compile-verified
//
#include <hip/hip_runtime.h>
#include <hip/hip_bf16.h>

typedef __attribute__((ext_vector_type(2))) float v2f;
typedef __attribute__((ext_vector_type(8))) float v8f;

#define NMOD   3
#define DD     30
#define LL     100
#define KK     20
#define NN     3000            // D*L
#define NT     9000            // 3*N
#define ND     256
#define PI_F   3.14159265358979323846f

// d_out layout (floats)
#define FE_OFF   0
#define ADJ_OFF  (NT*512)                    // 4,608,000
#define NB_OFF   (ADJ_OFF + (long)NT*NT)     // 85,608,000
#define FT_OFF   (NB_OFF + (long)NN*1536)    // 90,216,000

// scratch inside adjacency region of d_out (destroyed at the end by adj write)
#define SC_NORMED  0
#define SC_H0      (NT*ND)
#define SC_HA      (2*NT*ND)
#define SC_HS      (3*NT*ND)

// d_ws layout (floats)
#define WS_SPBLK   0                         // 90*100*100
#define WS_ANORM   900000
#define WS_DSIM    1800000                   // [3][3][3000]
#define WS_CROSSN  1827000                   // [3][3][3000]
#define WS_DINV    1854000                   // 9000

#define AS_STRIDE  260                       // 48-row A strip: 16B-aligned rows, conflict-free cols
#define AB_STRIDE  102                       // 100x100 block: 8B-aligned, conflict-free cols

__device__ __forceinline__ const float* xrow(const float* t, const float* v,
                                             const float* p, int g) {
    return g < NN ? t + (long)g*ND
                  : (g < 2*NN ? v + (long)(g-NN)*ND : p + (long)(g-2*NN)*ND);
}

// ---- 1. row-normalize t,v,p into normed (9000x256) ------------------------
__global__ void rownorm_kernel(const float* __restrict__ t, const float* __restrict__ v,
                               const float* __restrict__ p, float* __restrict__ normed) {
    int g = blockIdx.x*blockDim.x + threadIdx.x;
    if (g >= NT) return;
    const float* src = xrow(t, v, p, g);
    float s = 0.f;
    for (int k = 0; k < ND; ++k) s += src[k]*src[k];
    float inv = 1.0f / sqrtf(s);
    float* dst = normed + (long)g*ND;
    for (int k = 0; k < ND; ++k) dst[k] = src[k]*inv;
}

// ---- 2. per-(m,d) similarity blocks -> spital -----------------------------
__global__ __launch_bounds__(256)
void simblock_kernel(const float* __restrict__ normed, float* __restrict__ spblk) {
    int blk = blockIdx.x;                 // 0..89
    int m = blk / DD, d = blk % DD;
    const float* base = normed + (long)(m*NN + d*LL)*ND;
    for (int e = threadIdx.x; e < LL*LL; e += 256) {
        int i = e / LL, j = e % LL;
        const float* ri = base + (long)i*ND;
        const float* rj = base + (long)j*ND;
        float s = 0.f;
        for (int k = 0; k < ND; ++k) s += ri[k]*rj[k];
        float sim = 1.f - acosf(s*0.99999f)*(1.f/PI_F);
        float dist = fabsf((float)(i - j));
        float decay = logf(dist + 1.f) + 1e-4f;
        spblk[(long)blk*LL*LL + e] = sim/decay;
    }
}

// ---- 3. cross-modality per-node sims --------------------------------------
__global__ void crossdsim_kernel(const float* __restrict__ normed, float* __restrict__ dsim) {
    int id = blockIdx.x*blockDim.x + threadIdx.x;
    if (id >= 3*NN) return;
    int pr = id / NN, i = id % NN;
    int m = (pr == 2) ? 1 : 0;
    int n = (pr == 0) ? 1 : 2;
    const float* a = normed + (long)(m*NN + i)*ND;
    const float* b = normed + (long)(n*NN + i)*ND;
    float s = 0.f;
    for (int k = 0; k < ND; ++k) s += a[k]*b[k];
    float val = 1.f - acosf(s*0.99999f)*(1.f/PI_F);
    dsim[(m*3+n)*NN + i] = val;
    dsim[(n*3+m)*NN + i] = val;
}

// ---- 4. sequential sparsify (matches jax.lax.scan with stable argsort) ----
__global__ __launch_bounds__(128)
void sparsify_kernel(float* __restrict__ spblk) {
    __shared__ float A[LL*101];
    __shared__ float cbuf[LL];
    int tid = threadIdx.x;
    float* blk = spblk + (long)blockIdx.x*LL*LL;
    for (int e = tid; e < LL*LL; e += 128) A[(e/LL)*101 + (e%LL)] = blk[e];
    __syncthreads();
    for (int e = 0; e < LL; ++e) {
        // column e: zero all but top-K (ties kept at larger index, matching stable sort)
        if (tid < LL) cbuf[tid] = A[tid*101 + e];
        __syncthreads();
        if (tid < LL) {
            float x = cbuf[tid]; int cnt = 0;
            for (int j = 0; j < LL; ++j) {
                float vv = cbuf[j];
                cnt += (vv > x) || (vv == x && j > tid);
            }
            if (cnt >= KK) A[tid*101 + e] = 0.f;
        }
        __syncthreads();
        // row e (sees the column update, as in the reference)
        if (tid < LL) cbuf[tid] = A[e*101 + tid];
        __syncthreads();
        if (tid < LL) {
            float x = cbuf[tid]; int cnt = 0;
            for (int j = 0; j < LL; ++j) {
                float vv = cbuf[j];
                cnt += (vv > x) || (vv == x && j > tid);
            }
            if (cnt >= KK) A[e*101 + tid] = 0.f;
        }
        __syncthreads();
    }
    for (int e = tid; e < LL*LL; e += 128) blk[e] = A[(e/LL)*101 + (e%LL)];
}

// ---- 5. degrees -> dinv ----------------------------------------------------
__global__ void degree_kernel(const float* __restrict__ spblk, const float* __restrict__ dsim,
                              float* __restrict__ dinv) {
    int g = blockIdx.x*blockDim.x + threadIdx.x;
    if (g >= NT) return;
    int m = g / NN, i = g % NN, d = i / LL, r = i % LL;
    const float* row = spblk + (long)(m*DD + d)*LL*LL + r*LL;
    float s = 0.f;
    for (int j = 0; j < LL; ++j) s += row[j];
    for (int n = 0; n < 3; ++n) if (n != m) s += dsim[(m*3+n)*NN + i];
    if (s == 0.f) s = 1e-12f;
    dinv[g] = 1.0f / sqrtf(s);
}

// ---- 6. normalized blocks / cross values (kept in ws for SpMM + adj) ------
__global__ void blocknorm_kernel(const float* __restrict__ spblk, const float* __restrict__ dinv,
                                 float* __restrict__ anorm) {
    int e = blockIdx.x*blockDim.x + threadIdx.x;
    if (e >= 90*LL*LL) return;
    int blk = e / (LL*LL), rc = e % (LL*LL);
    int m = blk / DD, d = blk % DD;
    int gi = m*NN + d*LL + rc/LL;
    int gj = m*NN + d*LL + rc%LL;
    anorm[e] = dinv[gi]*spblk[e]*dinv[gj];
}

__global__ void crossnorm_kernel(const float* __restrict__ dsim, const float* __restrict__ dinv,
                                 float* __restrict__ crossn) {
    int id = blockIdx.x*blockDim.x + threadIdx.x;
    if (id >= 9*NN) return;
    int mn = id / NN, i = id % NN;
    int m = mn / 3, n = mn % 3;
    if (m == n) return;
    crossn[id] = dinv[m*NN + i]*dsim[id]*dinv[n*NN + i];
}

// ---- 7. 256-wide GEMM, fp32 WMMA, 48-row M-groups -------------------------
// mode 0: out = relu(A@W + bias)        (A = concat rows from a0/a1/a2)
// mode 1: out = relu(theta*(A@W) + (1-theta)*cin)
// Each wave: 2 n-tiles x 3 m-tiles; B fragment loaded once per (k, n-tile).
__global__ __launch_bounds__(256)
void gemm256_kernel(const float* __restrict__ a0, const float* __restrict__ a1,
                    const float* __restrict__ a2, const float* __restrict__ W,
                    const float* __restrict__ bias, const float* __restrict__ cin,
                    float* __restrict__ outp, float theta, int mode) {
    __shared__ float As[48*AS_STRIDE];
    int tid = threadIdx.x;
    int m0 = blockIdx.x*48;
    // stage 48x256 A strip as float4 (row pitch 260 floats = 1040B, 16B aligned)
    for (int c = tid; c < 48*64; c += 256) {
        int r = c >> 6, c4 = (c & 63) << 2;
        int row = m0 + r;
        float4 val = make_float4(0.f, 0.f, 0.f, 0.f);
        if (row < NT) val = *(const float4*)(xrow(a0, a1, a2, row) + c4);
        *(float4*)(&As[r*AS_STRIDE + c4]) = val;
    }
    __syncthreads();
    int wave = tid >> 5, lane = tid & 31;
    int lh = lane & 15, hi = lane >> 4;
    bool full = (m0 + 48 <= NT);            // block-uniform
    for (int nt = wave*2; nt < wave*2 + 2; ++nt) {
        int n0 = nt*16;
        int col = n0 + lh;
        v8f acc[3] = {};
        for (int k = 0; k < ND; k += 4) {
            int ka = k + hi*2;
            v2f b;
            b.x = W[(long)ka*ND + col];
            b.y = W[(long)(ka+1)*ND + col];
#pragma unroll
            for (int mt = 0; mt < 3; ++mt) {
                int ar = mt*16 + lh;
                v2f a;
                a.x = As[ar*AS_STRIDE + ka];
                a.y = As[ar*AS_STRIDE + ka + 1];
                acc[mt] = __builtin_amdgcn_wmma_f32_16x16x4_f32(false, a, false, b,
                                                                (short)0, acc[mt], false, false);
            }
        }
        float bcol = (mode == 0) ? bias[col] : 0.f;
        if (full) {
#pragma unroll
            for (int mt = 0; mt < 3; ++mt)
#pragma unroll
                for (int vi = 0; vi < 8; ++vi) {
                    int row = m0 + mt*16 + vi + hi*8;
                    float x = acc[mt][vi], o;
                    if (mode == 0) o = x + bcol;
                    else { float s = cin[(long)row*ND + col]; o = theta*x + (1.f - theta)*s; }
                    outp[(long)row*ND + col] = o > 0.f ? o : 0.f;
                }
        } else {
#pragma unroll
            for (int mt = 0; mt < 3; ++mt)
#pragma unroll
                for (int vi = 0; vi < 8; ++vi) {
                    int row = m0 + mt*16 + vi + hi*8;
                    if (row < NT) {
                        float x = acc[mt][vi], o;
                        if (mode == 0) o = x + bcol;
                        else { float s = cin[(long)row*ND + col]; o = theta*x + (1.f - theta)*s; }
                        outp[(long)row*ND + col] = o > 0.f ? o : 0.f;
                    }
                }
        }
    }
}

// ---- 8. structured SpMM: support = 0.9*(adj@h) + 0.1*h0 -------------------
// Each wave: 2 n-tiles, all 7 m-tiles (B fragment amortized 7x).
__global__ __launch_bounds__(256)
void spmm_support_kernel(const float* __restrict__ anorm, const float* __restrict__ crossn,
                         const float* __restrict__ h, const float* __restrict__ h0,
                         float* __restrict__ support) {
    __shared__ float Ab[LL*AB_STRIDE];
    int blk = blockIdx.x;
    int m = blk / DD, d = blk % DD;
    int g0 = m*NN + d*LL;
    int tid = threadIdx.x;
    for (int e = tid; e < LL*LL; e += 256)
        Ab[(e/LL)*AB_STRIDE + (e%LL)] = anorm[(long)blk*LL*LL + e];
    __syncthreads();
    int wave = tid >> 5, lane = tid & 31;
    int lh = lane & 15, hi = lane >> 4;
    for (int nt = wave*2; nt < wave*2 + 2; ++nt) {
        int n0 = nt*16;
        int col = n0 + lh;
        v8f acc[7] = {};
        for (int k = 0; k < LL; k += 4) {
            int ka = k + hi*2;
            v2f b;
            b.x = h[(long)(g0 + ka)*ND + col];
            b.y = h[(long)(g0 + ka + 1)*ND + col];
#pragma unroll
            for (int mt = 0; mt < 7; ++mt) {
                int ar = mt*16 + lh;
                v2f a;
                if (mt < 6) {
                    a.x = Ab[ar*AB_STRIDE + ka];
                    a.y = Ab[ar*AB_STRIDE + ka + 1];
                } else {
                    bool ok = ar < LL;
                    a.x = ok ? Ab[ar*AB_STRIDE + ka]     : 0.f;
                    a.y = ok ? Ab[ar*AB_STRIDE + ka + 1] : 0.f;
                }
                acc[mt] = __builtin_amdgcn_wmma_f32_16x16x4_f32(false, a, false, b,
                                                                (short)0, acc[mt], false, false);
            }
        }
#pragma unroll
        for (int mt = 0; mt < 7; ++mt)
#pragma unroll
            for (int vi = 0; vi < 8; ++vi) {
                int rl = mt*16 + vi + hi*8;
                if (mt < 6 || rl < LL) {
                    int i = d*LL + rl;
                    int g = m*NN + i;
                    float cross = 0.f;
#pragma unroll
                    for (int n = 0; n < 3; ++n) if (n != m)
                        cross += crossn[(m*3+n)*NN + i] * h[(long)(n*NN + i)*ND + col];
                    support[(long)g*ND + col] =
                        0.9f*(acc[mt][vi] + cross) + 0.1f*h0[(long)g*ND + col];
                }
            }
    }
}

// ---- 9. output assembly (float4 paths, all offsets 16B aligned) -----------
__global__ void fe_kernel(const float* __restrict__ t, const float* __restrict__ v,
                          const float* __restrict__ p, const float* __restrict__ hfin,
                          float* __restrict__ out) {
    long id4 = (long)blockIdx.x*blockDim.x + threadIdx.x;   // over NT*128 float4s
    if (id4 >= (long)NT*128) return;
    int g = (int)(id4 >> 7), c4 = (int)(id4 & 127);
    float4 val;
    if (c4 < 64) val = *(const float4*)(xrow(t, v, p, g) + (c4 << 2));
    else         val = *(const float4*)(hfin + (long)g*ND + ((c4 - 64) << 2));
    ((float4*)out)[id4] = val;
}

__global__ void zero4_kernel(float4* __restrict__ p, long n4) {
    long i = (long)blockIdx.x*blockDim.x + threadIdx.x;
    long stride = (long)gridDim.x*blockDim.x;
    float4 z = make_float4(0.f, 0.f, 0.f, 0.f);
    for (; i < n4; i += stride) p[i] = z;
}

__global__ void scatter_blocks_kernel(const float* __restrict__ anorm, float* __restrict__ adj) {
    int e = blockIdx.x*blockDim.x + threadIdx.x;
    if (e >= 90*LL*LL) return;
    int blk = e / (LL*LL), rc = e % (LL*LL);
    int m = blk / DD, d = blk % DD;
    int gi = m*NN + d*LL + rc/LL;
    int gj = m*NN + d*LL + rc%LL;
    adj[(long)gi*NT + gj] = anorm[e];
}

__global__ void scatter_cross_kernel(const float* __restrict__ crossn, float* __restrict__ adj) {
    int id = blockIdx.x*blockDim.x + threadIdx.x;
    if (id >= 9*NN) return;
    int mn = id / NN, i = id % NN;
    int m = mn / 3, n = mn % 3;
    if (m == n) return;
    adj[(long)(m*NN + i)*NT + (n*NN + i)] = crossn[id];
}

__global__ void nb_kernel(float* __restrict__ out) {
    long id4 = (long)blockIdx.x*blockDim.x + threadIdx.x;   // over NN*384 float4s
    if (id4 >= (long)NN*384) return;
    int i = (int)(id4 / 384), c4 = (int)(id4 % 384);
    int seg = c4 >> 7, cc4 = c4 & 127;
    float4 val = ((const float4*)out)[(long)(seg*NN + i)*128 + cc4];  // fe row = 128 float4
    ((float4*)(out + NB_OFF))[id4] = val;
}

__global__ void ftvp_kernel(float* __restrict__ out) {
    long id4 = (long)blockIdx.x*blockDim.x + threadIdx.x;   // over NT*128 float4s
    if (id4 >= (long)NT*128) return;
    ((float4*)(out + FT_OFF))[id4] = ((const float4*)out)[id4];  // ft|fv|fp == features_e
}

// ---------------------------------------------------------------------------
extern "C" void kernel_launch(void* const* d_in, const int* in_sizes, int n_in,
                              void* d_out, int out_size, void* d_ws, size_t ws_size,
                              hipStream_t stream) {
    const float* t  = (const float*)d_in[0];
    const float* v  = (const float*)d_in[1];
    const float* p  = (const float*)d_in[2];
    const float* W0 = (const float*)d_in[3];
    const float* b0 = (const float*)d_in[4];
    const float* Ws = (const float*)d_in[5];
    (void)in_sizes; (void)n_in; (void)out_size; (void)ws_size;

    float* out = (float*)d_out;
    float* adj = out + ADJ_OFF;
    // scratch inside adjacency region (overwritten by final adj fill)
    float* normed = adj + SC_NORMED;
    float* h0     = adj + SC_H0;
    float* hA     = adj + SC_HA;
    float* hS     = adj + SC_HS;

    float* wsf    = (float*)d_ws;
    float* spblk  = wsf + WS_SPBLK;
    float* anorm  = wsf + WS_ANORM;
    float* dsim   = wsf + WS_DSIM;
    float* crossn = wsf + WS_CROSSN;
    float* dinv   = wsf + WS_DINV;

    const float TH0 = 0.40546510810816444f;   // log(0.5/1 + 1)
    const float TH1 = 0.22314355131420976f;   // log(0.5/2 + 1)

    // adjacency construction
    rownorm_kernel  <<<(NT+255)/256, 256, 0, stream>>>(t, v, p, normed);
    simblock_kernel <<<90, 256, 0, stream>>>(normed, spblk);
    crossdsim_kernel<<<(3*NN+255)/256, 256, 0, stream>>>(normed, dsim);
    sparsify_kernel <<<90, 128, 0, stream>>>(spblk);
    degree_kernel   <<<(NT+255)/256, 256, 0, stream>>>(spblk, dsim, dinv);
    blocknorm_kernel<<<(90*LL*LL+255)/256, 256, 0, stream>>>(spblk, dinv, anorm);
    crossnorm_kernel<<<(9*NN+255)/256, 256, 0, stream>>>(dsim, dinv, crossn);

    // GCN-II (structure-exploiting: adj@h as 90 block GEMMs + diagonal cross terms)
    int gblocks = (NT + 47)/48;   // 188
    gemm256_kernel<<<gblocks, 256, 0, stream>>>(t, v, p, W0, b0, nullptr, h0, 0.f, 0);
    // layer 0
    spmm_support_kernel<<<90, 256, 0, stream>>>(anorm, crossn, h0, h0, hS);
    gemm256_kernel<<<gblocks, 256, 0, stream>>>(hS, hS + (long)NN*ND, hS + (long)2*NN*ND,
                                                Ws, nullptr, hS, hA, TH0, 1);
    // layer 1
    spmm_support_kernel<<<90, 256, 0, stream>>>(anorm, crossn, hA, h0, hS);
    gemm256_kernel<<<gblocks, 256, 0, stream>>>(hS, hS + (long)NN*ND, hS + (long)2*NN*ND,
                                                Ws + (long)ND*ND, nullptr, hS, hA, TH1, 1);

    // features_e (before the adj region scratch is destroyed)
    fe_kernel<<<(int)(((long)NT*128 + 255)/256), 256, 0, stream>>>(t, v, p, hA, out);

    // dense adjacency: zero-fill + scatter structural nonzeros (dominant HBM traffic)
    zero4_kernel<<<8192, 256, 0, stream>>>((float4*)adj, (long)NT*NT/4);
    scatter_blocks_kernel<<<(90*LL*LL+255)/256, 256, 0, stream>>>(anorm, adj);
    scatter_cross_kernel <<<(9*NN+255)/256, 256, 0, stream>>>(crossn, adj);

    // neighbor_emb, ft/fv/fp (read from features_e region of d_out)
    nb_kernel  <<<(int)(((long)NN*384 + 255)/256), 256, 0, stream>>>(out);
    ftvp_kernel<<<(int)(((long)NT*128 + 255)/256), 256, 0, stream>>>(out);
}